// Node_AE_14499809591358
// MI455X (gfx1250) — compile-verified
//
#include <hip/hip_runtime.h>
#include <math.h>

#define N_NODES 8192
#define N_EDGES 262144
#define IN_NODE 11
#define IN_EDGE 4
#define IN_NF   15   // IN_NODE + IN_EDGE
#define H_NF    4
#define OUT_NF  4
#define EMB_NF  2

typedef float v2f __attribute__((ext_vector_type(2)));
typedef float v8f __attribute__((ext_vector_type(8)));

// -------- workspace layout (in floats) --------
//  [0,          N*4)        agg      (segment-sum result)
//  [N*4,        N*6)        emb      (node embeddings, interleaved x,y)
//  [N*6,        N*7)        nrm      (|emb|^2 per node)
//  [N*7,        N*7+30)     Wc       (folded 15x2 weight, row-major)
//  [N*7+32,     N*7+34)     bc       (folded bias)
#define AGG_OFF 0
#define EMB_OFF (N_NODES * 4)
#define NRM_OFF (N_NODES * 6)
#define WC_OFF  (N_NODES * 7)
#define BC_OFF  (N_NODES * 7 + 32)

// ---- zero the aggregation buffer ----
__global__ void k_zero(float* __restrict__ agg) {
    int i = blockIdx.x * blockDim.x + threadIdx.x;
    agg[i] = 0.0f;   // launched with exactly N*4 threads
}

// ---- segment_sum(edge_attr, row) via fp32 global atomics ----
__global__ void k_scatter(const int* __restrict__ eidx,
                          const float* __restrict__ eattr,
                          float* __restrict__ agg) {
    int e = blockIdx.x * blockDim.x + threadIdx.x;
    if (e >= N_EDGES) return;
    int r = eidx[e];                       // edge_index[0][e]
    const float* a = eattr + (size_t)e * IN_EDGE;
    float* dst = agg + (size_t)r * IN_EDGE;
    atomicAdd(dst + 0, a[0]);
    atomicAdd(dst + 1, a[1]);
    atomicAdd(dst + 2, a[2]);
    atomicAdd(dst + 3, a[3]);
}

// ---- fold the purely-affine MLP:  Wc = W1@W2@We,  bc = (b1@W2 + b2)@We + be ----
__global__ void k_fold(const float* __restrict__ W1, const float* __restrict__ b1,
                       const float* __restrict__ W2, const float* __restrict__ b2,
                       const float* __restrict__ We, const float* __restrict__ be,
                       float* __restrict__ wc, float* __restrict__ bc) {
    if (threadIdx.x != 0 || blockIdx.x != 0) return;
    float W12[IN_NF][H_NF];
    for (int i = 0; i < IN_NF; ++i)
        for (int j = 0; j < OUT_NF; ++j) {
            float s = 0.0f;
            for (int k = 0; k < H_NF; ++k) s += W1[i * H_NF + k] * W2[k * OUT_NF + j];
            W12[i][j] = s;
        }
    float b12[OUT_NF];
    for (int j = 0; j < OUT_NF; ++j) {
        float s = b2[j];
        for (int k = 0; k < H_NF; ++k) s += b1[k] * W2[k * OUT_NF + j];
        b12[j] = s;
    }
    for (int i = 0; i < IN_NF; ++i)
        for (int j2 = 0; j2 < EMB_NF; ++j2) {
            float s = 0.0f;
            for (int j = 0; j < OUT_NF; ++j) s += W12[i][j] * We[j * EMB_NF + j2];
            wc[i * EMB_NF + j2] = s;
        }
    for (int j2 = 0; j2 < EMB_NF; ++j2) {
        float s = be[j2];
        for (int j = 0; j < OUT_NF; ++j) s += b12[j] * We[j * EMB_NF + j2];
        bc[j2] = s;
    }
}

// ---- per-node embedding + squared-norm; also write node_emb output tail ----
__global__ void k_nodes(const float* __restrict__ nf, const float* __restrict__ agg,
                        const float* __restrict__ wc, const float* __restrict__ bc,
                        float* __restrict__ emb, float* __restrict__ nrm,
                        float* __restrict__ out_tail) {
    int i = blockIdx.x * blockDim.x + threadIdx.x;
    if (i >= N_NODES) return;
    float in[IN_NF];
#pragma unroll
    for (int t = 0; t < IN_NODE; ++t) in[t] = nf[(size_t)i * IN_NODE + t];
#pragma unroll
    for (int t = 0; t < IN_EDGE; ++t) in[IN_NODE + t] = agg[(size_t)i * IN_EDGE + t];
    float x = bc[0], y = bc[1];
#pragma unroll
    for (int t = 0; t < IN_NF; ++t) {
        x = fmaf(in[t], wc[t * 2 + 0], x);
        y = fmaf(in[t], wc[t * 2 + 1], y);
    }
    emb[2 * i]     = x;
    emb[2 * i + 1] = y;
    nrm[i] = x * x + y * y;
    out_tail[2 * i]     = x;   // node_emb output (second tuple element)
    out_tail[2 * i + 1] = y;
}

// ---- all-pairs: one wave32 computes a 16x16 tile via V_WMMA_F32_16X16X4_F32 ----
// Gram tile G = A(16x4) x B(4x16), K slots 0,1 = (x,y), slots 2,3 = 0.
// d_ij = nrm_i + nrm_j - 2*G_ij ; out = sigmoid(10 d - 1), diag zeroed.
__global__ __launch_bounds__(256) void k_adj(const float* __restrict__ emb,
                                             const float* __restrict__ nrm,
                                             float* __restrict__ out) {
    const int wave = threadIdx.x >> 5;
    const int lane = threadIdx.x & 31;
    const int tile = blockIdx.x * 8 + wave;       // 512x512 tiles total
    const int rowTile = tile >> 9;                // tile / 512
    const int colTile = tile & 511;               // tile % 512
    const int row0 = rowTile << 4;
    const int col0 = colTile << 4;
    const int lo = lane & 15;
    const int hi = lane >> 4;

    const float2* e2 = (const float2*)emb;
    float2 ea = e2[row0 + lo];                    // row embeddings (A, M = lo)
    float2 eb = e2[col0 + lo];                    // col embeddings (B, N = lo)

    v2f A; A.x = hi ? 0.0f : ea.x; A.y = hi ? 0.0f : ea.y;  // K=0 -> x, K=1 -> y
    v2f B; B.x = hi ? 0.0f : eb.x; B.y = hi ? 0.0f : eb.y;
    v8f C = {};
    C = __builtin_amdgcn_wmma_f32_16x16x4_f32(
            /*neg_a=*/false, A, /*neg_b=*/false, B,
            /*c_mod=*/(short)0, C, /*reuse_a=*/false, /*reuse_b=*/false);

    const int n = col0 + lo;                      // global column for this lane
    const float nrm_n = nrm[n];
    const int rbase = row0 + (hi << 3);           // lanes 16-31 hold rows M=8..15
#pragma unroll
    for (int v = 0; v < 8; ++v) {
        const int m = rbase + v;
        float g = C[v];
        float d = nrm[m] + nrm_n - 2.0f * g;
        // sigmoid(10d - 1) = 1 / (1 + exp(1 - 10d)); fast v_rcp_f32 (~1 ulp) is
        // plenty for a sigmoid output -- avoids the IEEE v_div_scale sequence.
        float s = __builtin_amdgcn_rcpf(1.0f + __expf(1.0f - 10.0f * d));
        if (m == n) s = 0.0f;
        __builtin_nontemporal_store(s, out + (size_t)m * N_NODES + n);
    }
}

extern "C" void kernel_launch(void* const* d_in, const int* in_sizes, int n_in,
                              void* d_out, int out_size, void* d_ws, size_t ws_size,
                              hipStream_t stream) {
    const float* node_feats = (const float*)d_in[0];
    const int*   edge_index = (const int*)d_in[1];   // row = first E entries
    const float* edge_attr  = (const float*)d_in[2];
    const float* W1 = (const float*)d_in[3];
    const float* b1 = (const float*)d_in[4];
    const float* W2 = (const float*)d_in[5];
    const float* b2 = (const float*)d_in[6];
    const float* We = (const float*)d_in[7];
    const float* be = (const float*)d_in[8];

    float* ws  = (float*)d_ws;
    float* agg = ws + AGG_OFF;
    float* emb = ws + EMB_OFF;
    float* nrm = ws + NRM_OFF;
    float* wc  = ws + WC_OFF;
    float* bc  = ws + BC_OFF;

    float* out      = (float*)d_out;                 // [N,N] adjacency
    float* out_tail = out + (size_t)N_NODES * N_NODES;  // [N,2] node_emb

    k_zero<<<(N_NODES * 4) / 256, 256, 0, stream>>>(agg);
    k_scatter<<<N_EDGES / 256, 256, 0, stream>>>(edge_index, edge_attr, agg);
    k_fold<<<1, 32, 0, stream>>>(W1, b1, W2, b2, We, be, wc, bc);
    k_nodes<<<N_NODES / 256, 256, 0, stream>>>(node_feats, agg, wc, bc, emb, nrm, out_tail);

    const int totalTiles = (N_NODES / 16) * (N_NODES / 16);  // 262144
    k_adj<<<totalTiles / 8, 256, 0, stream>>>(emb, nrm, out);
}